// GATLayer_6502580486178
// MI455X (gfx1250) — compile-verified
//
#include <hip/hip_runtime.h>
#include <math.h>

// GAT layer for MI455X (gfx1250, wave32).
// k1: h = X@W + bias via V_WMMA_F32_16X16X4_F32, fused wh1/wh2 epilogue.
// k2: per-node (wave32) softmax over 16 edges + attention-weighted gather of h rows.

typedef float v2f __attribute__((ext_vector_type(2)));
typedef float v8f __attribute__((ext_vector_type(8)));

#define GAT_ALPHA 0.2f
#define GAT_EPS   1e-12f

__global__ __launch_bounds__(256) void gat_gemm_wmma(
    const float* __restrict__ X,     // N x 128 row-major
    const float* __restrict__ W,     // 128 x 64 row-major
    const float* __restrict__ bias,  // 64
    const float* __restrict__ a,     // 128 (a_src = a[0:64], a_dst = a[64:128])
    float* __restrict__ H,           // N x 64 (padded to mult of 16 rows)
    float* __restrict__ wh1,         // N
    float* __restrict__ wh2,         // N
    int n)
{
    const int gwave = (int)((blockIdx.x * blockDim.x + threadIdx.x) >> 5);
    const int lane  = (int)(threadIdx.x & 31);
    const int row0  = gwave * 16;
    if (row0 >= n) return;             // whole wave exits together; EXEC all-1s for WMMA

    const int half = lane >> 4;        // 0: K pair {0,1} / rows 0-7 ; 1: K pair {2,3} / rows 8-15
    const int idx  = lane & 15;        // M row within tile (A), N col within tile (B/C/D)

    int ra = row0 + idx;
    if (ra >= n) ra = n - 1;           // clamp reads for a partial last tile
    const float* xrow = X + (size_t)ra * 128 + 2 * half;

    v8f acc0 = {}; v8f acc1 = {}; v8f acc2 = {}; v8f acc3 = {};

    #pragma unroll 4
    for (int k = 0; k < 128; k += 4) {
        v2f av;
        av.x = xrow[k];
        av.y = xrow[k + 1];
        const float* w0 = W + (size_t)(k + 2 * half) * 64 + idx;
        const float* w1 = w0 + 64;
        v2f b0; b0.x = w0[0];  b0.y = w1[0];
        v2f b1; b1.x = w0[16]; b1.y = w1[16];
        v2f b2; b2.x = w0[32]; b2.y = w1[32];
        v2f b3; b3.x = w0[48]; b3.y = w1[48];
        acc0 = __builtin_amdgcn_wmma_f32_16x16x4_f32(false, av, false, b0, (short)0, acc0, false, false);
        acc1 = __builtin_amdgcn_wmma_f32_16x16x4_f32(false, av, false, b1, (short)0, acc1, false, false);
        acc2 = __builtin_amdgcn_wmma_f32_16x16x4_f32(false, av, false, b2, (short)0, acc2, false, false);
        acc3 = __builtin_amdgcn_wmma_f32_16x16x4_f32(false, av, false, b3, (short)0, acc3, false, false);
    }

    // Epilogue: bias add, store h, fused wh1/wh2 dot products (cross-lane reduce within 16-lane half)
    const float bc0 = bias[idx], bc1 = bias[16 + idx], bc2 = bias[32 + idx], bc3 = bias[48 + idx];
    const float as0 = a[idx],      as1 = a[16 + idx],  as2 = a[32 + idx],  as3 = a[48 + idx];
    const float ad0 = a[64 + idx], ad1 = a[80 + idx],  ad2 = a[96 + idx],  ad3 = a[112 + idx];

    #pragma unroll
    for (int r = 0; r < 8; ++r) {
        const int row = row0 + r + half * 8;   // D layout: VGPR r -> rows r (lanes 0-15), r+8 (lanes 16-31)
        const float h0 = acc0[r] + bc0;
        const float h1 = acc1[r] + bc1;
        const float h2 = acc2[r] + bc2;
        const float h3 = acc3[r] + bc3;
        float* hrow = H + (size_t)row * 64;
        hrow[idx]      = h0;
        hrow[16 + idx] = h1;
        hrow[32 + idx] = h2;
        hrow[48 + idx] = h3;

        float p1 = h0 * as0 + h1 * as1 + h2 * as2 + h3 * as3;
        float p2 = h0 * ad0 + h1 * ad1 + h2 * ad2 + h3 * ad3;
        #pragma unroll
        for (int s = 8; s >= 1; s >>= 1) {
            p1 += __shfl_xor(p1, s, 16);
            p2 += __shfl_xor(p2, s, 16);
        }
        if (idx == 0 && row < n) { wh1[row] = p1; wh2[row] = p2; }
    }
}

__global__ __launch_bounds__(256) void gat_aggregate(
    const float* __restrict__ H,
    const float* __restrict__ wh1,
    const float* __restrict__ wh2,
    const int*   __restrict__ indptr,
    const int*   __restrict__ indices,
    float* __restrict__ out,
    int n)
{
    const int i    = (int)((blockIdx.x * blockDim.x + threadIdx.x) >> 5);
    const int lane = (int)(threadIdx.x & 31);
    if (i >= n) return;

    const int start = indptr[i];
    int deg = indptr[i + 1] - start;
    if (deg > 16) deg = 16;            // fixed degree 16 in this problem

    const int j = lane & 15;           // both lane halves mirror the edge work -> free broadcast
    const int col = indices[start + ((j < deg) ? j : 0)];

    float e = wh1[i] + wh2[col];
    e = (e > 0.0f) ? e : GAT_ALPHA * e;
    if (j >= deg) e = -__builtin_inff();

    float m = e;
    #pragma unroll
    for (int s = 8; s >= 1; s >>= 1) m = fmaxf(m, __shfl_xor(m, s, 16));

    const float ex = (j < deg) ? __expf(e - m) : 0.0f;
    float ssum = ex;
    #pragma unroll
    for (int s = 8; s >= 1; s >>= 1) ssum += __shfl_xor(ssum, s, 16);

    const float attn = ex / (ssum + GAT_EPS);

    // 64 features across 32 lanes as float2; per-edge 256B coalesced row read (L2-resident h)
    float accx = 0.0f, accy = 0.0f;
    for (int jj = 0; jj < deg; ++jj) {
        const float aj = __shfl(attn, jj, 16);
        const int   cj = __shfl(col,  jj, 16);
        const float2* hp = (const float2*)(H + (size_t)cj * 64);
        const float2 hv = hp[lane];
        accx = fmaf(aj, hv.x, accx);
        accy = fmaf(aj, hv.y, accy);
    }
    float2* orow = (float2*)(out + (size_t)i * 64);
    orow[lane] = make_float2(accx, accy);
}

extern "C" void kernel_launch(void* const* d_in, const int* in_sizes, int n_in,
                              void* d_out, int out_size, void* d_ws, size_t ws_size,
                              hipStream_t stream) {
    const float* X       = (const float*)d_in[0];  // input_h: N x 128
    const float* W       = (const float*)d_in[1];  // 128 x 64
    const float* a       = (const float*)d_in[2];  // 128 x 1
    const float* bias    = (const float*)d_in[3];  // 64
    const int*   indptr  = (const int*)  d_in[4];  // N+1
    const int*   indices = (const int*)  d_in[5];  // E
    float* out = (float*)d_out;

    const int n    = in_sizes[0] / 128;
    const int npad = (n + 15) & ~15;

    float* H   = (float*)d_ws;                     // npad * 64 floats
    float* wh1 = H + (size_t)npad * 64;            // npad floats
    float* wh2 = wh1 + npad;                       // npad floats

    const int nwaves1  = (n + 15) / 16;
    const int threads1 = nwaves1 * 32;
    dim3 blk(256);
    dim3 grd1((unsigned)((threads1 + 255) / 256));
    hipLaunchKernelGGL(gat_gemm_wmma, grd1, blk, 0, stream, X, W, bias, a, H, wh1, wh2, n);

    const long long threads2 = (long long)n * 32;
    dim3 grd2((unsigned)((threads2 + 255) / 256));
    hipLaunchKernelGGL(gat_aggregate, grd2, blk, 0, stream, H, wh1, wh2, indptr, indices, out, n);
}